// knnLoss_37890201485849
// MI455X (gfx1250) — compile-verified
//
#include <hip/hip_runtime.h>

typedef __attribute__((ext_vector_type(2))) float v2f;
typedef __attribute__((ext_vector_type(8))) float v8f;

#define BATCH   4
#define NPTS    10000      // 100*100 downsampled points
#define NTILES  625        // NPTS / 16
#define PLANE   (400*400)

// ---------------------------------------------------------------------------
// Prep: gather strided pixels (B,3,400,400), every 4th row/col, into
// WMMA-native per-lane layouts.
//   SA[(b*625+tile)*32+lane] = A-matrix float2: lanes0-15 (sx,sy), lanes16-31 (sz,|s|^2)
//   TB[...]                  = B-matrix float2: lanes0-15 (-2tx,-2ty), lanes16-31 (-2tz, 1)
//   TC[...]                  = C broadcast: |t_n|^2 for n = tile*16 + (lane&15)
// ---------------------------------------------------------------------------
__global__ void knn_prep_kernel(const float* __restrict__ tpc,
                                const float* __restrict__ spc,
                                float* __restrict__ SA,
                                float* __restrict__ TB,
                                float* __restrict__ TC) {
    int idx = blockIdx.x * blockDim.x + threadIdx.x;
    if (idx >= BATCH * NTILES * 32) return;
    int lane = idx & 31;
    int tile = (idx >> 5) % NTILES;
    int b    = idx / (NTILES * 32);
    int r    = lane & 15;
    int half = lane >> 4;

    int n = tile * 16 + r;           // point index 0..9999
    int i = n / 100;                 // downsampled row
    int j = n % 100;                 // downsampled col
    size_t base = (size_t)b * 3 * PLANE + (size_t)(4 * i) * 400 + (size_t)(4 * j);

    float tx = tpc[base], ty = tpc[base + PLANE], tz = tpc[base + 2 * (size_t)PLANE];
    float sx = spc[base], sy = spc[base + PLANE], sz = spc[base + 2 * (size_t)PLANE];
    float tn2 = tx * tx + ty * ty + tz * tz;
    float sn2 = sx * sx + sy * sy + sz * sz;

    size_t o = (size_t)(b * NTILES + tile) * 32 + lane;
    // A: K=0/1 in lanes 0-15, K=2/3 in lanes 16-31 (ISA 32-bit A 16x4 layout)
    SA[2 * o + 0] = half ? sz  : sx;
    SA[2 * o + 1] = half ? sn2 : sy;
    // B: K=0/1 rows in lanes 0-15, K=2/3 rows in lanes 16-31
    TB[2 * o + 0] = half ? (-2.0f * tz) : (-2.0f * tx);
    TB[2 * o + 1] = half ? 1.0f         : (-2.0f * ty);
    TC[o] = tn2;
}

// ---------------------------------------------------------------------------
// Hot loop: one wave per 16-target tile. 625 WMMA 16x16x4 f32 issues per wave,
// running per-lane min over the 8 accumulator VGPRs (8 sources per lane).
// ---------------------------------------------------------------------------
__global__ __launch_bounds__(128) void knn_min_kernel(const float* __restrict__ SA,
                                                      const float* __restrict__ TB,
                                                      const float* __restrict__ TC,
                                                      float* __restrict__ partial) {
    int lane    = threadIdx.x & 31;
    int wave    = threadIdx.x >> 5;
    int tileIdx = blockIdx.x * 4 + wave;          // 0 .. 2499, always in range
    int b  = tileIdx / NTILES;
    int nt = tileIdx % NTILES;

    const v2f* SA2 = (const v2f*)SA;
    const v2f* TB2 = (const v2f*)TB;

    size_t to = (size_t)(b * NTILES + nt) * 32 + lane;
    v2f bv   = TB2[to];           // B-matrix operand (fixed across inner loop)
    float cv = TC[to];            // |t|^2 broadcast for this lane's target column

    v8f cvec;
    v8f dmin;
    #pragma unroll
    for (int k = 0; k < 8; ++k) { cvec[k] = cv; dmin[k] = 3.4e38f; }

    size_t abase = (size_t)b * NTILES * 32;
    #pragma unroll 5
    for (int mt = 0; mt < NTILES; ++mt) {
        v2f av = SA2[abase + (size_t)mt * 32 + lane];
        // d[m][n] = |t_n|^2 - 2 s_m . t_n + |s_m|^2  (full squared distance)
        v8f d = __builtin_amdgcn_wmma_f32_16x16x4_f32(
            /*neg_a=*/false, av, /*neg_b=*/false, bv,
            /*c_mod=*/(short)0, cvec, /*reuse_a=*/false, /*reuse_b=*/false);
        #pragma unroll
        for (int k = 0; k < 8; ++k) dmin[k] = fminf(dmin[k], d[k]);
    }

    // per-lane min over 8 sources, then merge lane halves (M=0-7 vs M=8-15)
    float m = dmin[0];
    #pragma unroll
    for (int k = 1; k < 8; ++k) m = fminf(m, dmin[k]);
    m = fminf(m, __shfl_xor(m, 16, 32));
    m = fmaxf(m, 0.0f);                // reference distance is a true square >= 0

    // sum the 16 targets (lanes 0-15 and 16-31 hold duplicate target groups)
    m += __shfl_xor(m, 1, 32);
    m += __shfl_xor(m, 2, 32);
    m += __shfl_xor(m, 4, 32);
    m += __shfl_xor(m, 8, 32);
    if (lane == 0) partial[tileIdx] = m;
}

// ---------------------------------------------------------------------------
// Deterministic final reduction: single wave, fixed strided order + butterfly.
// out = (sum of per-target min d^2) / (B * 3 * N) = total / 120000
// ---------------------------------------------------------------------------
__global__ void knn_finalize_kernel(const float* __restrict__ partial,
                                    float* __restrict__ out) {
    int lane = threadIdx.x;    // 32 threads, 1 block
    float s = 0.0f;
    for (int i = lane; i < BATCH * NTILES; i += 32) s += partial[i];
    s += __shfl_xor(s, 1, 32);
    s += __shfl_xor(s, 2, 32);
    s += __shfl_xor(s, 4, 32);
    s += __shfl_xor(s, 8, 32);
    s += __shfl_xor(s, 16, 32);
    if (lane == 0) out[0] = s * (1.0f / (float)(BATCH * 3 * NPTS));
}

extern "C" void kernel_launch(void* const* d_in, const int* in_sizes, int n_in,
                              void* d_out, int out_size, void* d_ws, size_t ws_size,
                              hipStream_t stream) {
    (void)in_sizes; (void)n_in; (void)out_size; (void)ws_size;
    const float* tpc = (const float*)d_in[0];   // target_pc (4,3,400,400) f32
    const float* spc = (const float*)d_in[1];   // source_pc (4,3,400,400) f32

    float* ws      = (float*)d_ws;
    float* SA      = ws;                         // 4*625*32*2 = 160000 floats
    float* TB      = ws + 160000;                // 160000 floats
    float* TC      = ws + 320000;                //  80000 floats
    float* partial = ws + 400000;                //   2500 floats (~1.61 MB total)

    int prepThreads = BATCH * NTILES * 32;       // 80000
    knn_prep_kernel<<<(prepThreads + 255) / 256, 256, 0, stream>>>(tpc, spc, SA, TB, TC);
    knn_min_kernel<<<NTILES, 128, 0, stream>>>(SA, TB, TC, partial);   // 2500 waves
    knn_finalize_kernel<<<1, 32, 0, stream>>>(partial, (float*)d_out);
}